// SRR_63299228009149
// MI455X (gfx1250) — compile-verified
//
#include <hip/hip_runtime.h>
#include <math.h>

// ---------------------------------------------------------------------------
// GraphTransformer layer for MI455X (gfx1250, wave32).
// fp32 end-to-end (workload is HBM-bound: ~185 GFLOP vs ~4-5 GB traffic at
// 23.3 TB/s, so low-precision WMMA buys nothing; f32 WMMA keeps accuracy and
// still runs all GEMMs on the matrix pipes).
// Matrix ops: V_WMMA_F32_16X16X4_F32 via __builtin_amdgcn_wmma_f32_16x16x4_f32.
// ---------------------------------------------------------------------------

#define N_NODES 50000
#define N_EDGES 800000
#define DIM     128
#define NHEAD   8

typedef __attribute__((ext_vector_type(2))) float v2f;
typedef __attribute__((ext_vector_type(8))) float v8f;

// D = A(16x4) * B(4x16) + C(16x16), f32, wave32.
// A layout: lanes 0-15 = rows 0-15 with {K=k0,k0+1} in {x,y}; lanes 16-31 same
// rows with {K=k0+2,k0+3}.  B symmetric with N across lanes.  C/D: VGPR r ->
// row r (lanes 0-15) / row 8+r (lanes 16-31), col = lane&15.
__device__ __forceinline__ v8f wmma_f32(v2f a, v2f b, v8f c) {
    return __builtin_amdgcn_wmma_f32_16x16x4_f32(
        /*neg_a=*/false, a, /*neg_b=*/false, b,
        /*c_mod=*/(short)0, c, /*reuse_a=*/false, /*reuse_b=*/false);
}

// ------------------------------ utility ------------------------------------

__global__ void zero_f32(float* __restrict__ p, long n) {
    long i = (long)blockIdx.x * blockDim.x + threadIdx.x;
    long s = (long)gridDim.x * blockDim.x;
    for (; i < n; i += s) p[i] = 0.0f;
}

// ------------------------------ batchnorm ----------------------------------

// Column-wise sum / sum-of-squares over [M,128] row-major. 128 threads/block,
// one row per block iteration (fully coalesced); atomic finish.
__global__ void bn_stats(const float* __restrict__ x, long M,
                         float* __restrict__ st /* [0:128]=sum [128:256]=sumsq */) {
    const int col = threadIdx.x;            // 0..127
    float s = 0.0f, s2 = 0.0f;
    for (long r = blockIdx.x; r < M; r += gridDim.x) {
        float v = x[r * DIM + col];
        s += v; s2 += v * v;
    }
    atomicAdd(&st[col], s);
    atomicAdd(&st[DIM + col], s2);
}

// y = (x - mean) * rsqrt(var + eps) * g + b   (biased var, matches jnp.var)
__global__ void bn_apply(const float* __restrict__ x, const float* __restrict__ st,
                         const float* __restrict__ g, const float* __restrict__ b,
                         float* __restrict__ y, long M) {
    long i = (long)blockIdx.x * blockDim.x + threadIdx.x;
    long n = M * DIM;
    long s = (long)gridDim.x * blockDim.x;
    float invM = 1.0f / (float)M;
    for (; i < n; i += s) {
        int c = (int)(i & (DIM - 1));
        float mean = st[c] * invM;
        float var  = st[DIM + c] * invM - mean * mean;
        float rstd = rsqrtf(var + 1e-5f);
        y[i] = (x[i] - mean) * rstd * g[c] + b[c];
    }
}

// ------------------------------ WMMA GEMM ----------------------------------

// C[M,128] = A[M,128] @ W[128,128] (+bias)(+residual). M % 16 == 0.
// Block = 256 threads = 8 waves; wave w owns the 16x16 tile at cols w*16.
__global__ void gemm128_wmma(const float* __restrict__ A, const float* __restrict__ W,
                             const float* __restrict__ bias, const float* __restrict__ res,
                             float* __restrict__ C, int M) {
    const int lane = threadIdx.x & 31;
    const int wave = threadIdx.x >> 5;
    const int row0 = blockIdx.x * 16;
    const int col0 = wave * 16;
    const int half = lane >> 4;   // 0: K pair {0,1}, 1: K pair {2,3}
    const int l16  = lane & 15;

    v8f acc = {};
    const float* ap = A + (size_t)(row0 + l16) * DIM + half * 2;
    const float* bp = W + (size_t)(half * 2) * DIM + (col0 + l16);
    #pragma unroll 8
    for (int k0 = 0; k0 < DIM; k0 += 4) {
        v2f a, b;
        a.x = ap[0];      a.y = ap[1];
        b.x = bp[0];      b.y = bp[DIM];
        acc = wmma_f32(a, b, acc);
        ap += 4;
        bp += 4 * DIM;
    }

    const int crow = row0 + half * 8;
    const int ccol = col0 + l16;
    float bi = bias ? bias[ccol] : 0.0f;
    #pragma unroll
    for (int r = 0; r < 8; ++r) {
        float v = acc[r] + bi;
        size_t idx = (size_t)(crow + r) * DIM + ccol;
        if (res) v += res[idx];
        C[idx] = v;
    }
}

// Fused FFN: OUT = res + relu(X@W1 + b1) @ W2 + b2.
// X:[M,128] W1:[128,256] W2:[256,128]. 16-row tile, intermediate in LDS
// (stride 257 to avoid 16-way bank conflicts on column reads).
__global__ void ffn_wmma(const float* __restrict__ X,  const float* __restrict__ W1,
                         const float* __restrict__ b1, const float* __restrict__ W2,
                         const float* __restrict__ b2, const float* __restrict__ res,
                         float* __restrict__ OUT, int M) {
    __shared__ float t[16][257];
    const int lane = threadIdx.x & 31;
    const int wave = threadIdx.x >> 5;
    const int row0 = blockIdx.x * 16;
    const int half = lane >> 4;
    const int l16  = lane & 15;

    // ---- stage 1: t = relu(X[16,128] @ W1[128,256] + b1); wave w -> cols [w*32, w*32+32)
    #pragma unroll
    for (int sub = 0; sub < 2; ++sub) {
        const int col0 = wave * 32 + sub * 16;
        v8f acc = {};
        const float* ap = X  + (size_t)(row0 + l16) * DIM + half * 2;
        const float* bp = W1 + (size_t)(half * 2) * 256 + (col0 + l16);
        #pragma unroll 8
        for (int k0 = 0; k0 < DIM; k0 += 4) {
            v2f a, b;
            a.x = ap[0]; a.y = ap[1];
            b.x = bp[0]; b.y = bp[256];
            acc = wmma_f32(a, b, acc);
            ap += 4;
            bp += 4 * 256;
        }
        const int trow = half * 8;
        const int tcol = col0 + l16;
        float bi = b1[tcol];
        #pragma unroll
        for (int r = 0; r < 8; ++r)
            t[trow + r][tcol] = fmaxf(acc[r] + bi, 0.0f);
    }
    __syncthreads();

    // ---- stage 2: OUT = t[16,256] @ W2[256,128] + b2 + res; wave w -> cols [w*16, w*16+16)
    {
        const int col0 = wave * 16;
        v8f acc = {};
        const float* bp = W2 + (size_t)(half * 2) * DIM + (col0 + l16);
        #pragma unroll 8
        for (int k0 = 0; k0 < 256; k0 += 4) {
            int ka = k0 + half * 2;
            v2f a, b;
            a.x = t[l16][ka]; a.y = t[l16][ka + 1];
            b.x = bp[0];      b.y = bp[DIM];
            acc = wmma_f32(a, b, acc);
            bp += 4 * DIM;
        }
        const int crow = row0 + half * 8;
        const int ccol = col0 + l16;
        float bi = b2[ccol];
        #pragma unroll
        for (int r = 0; r < 8; ++r) {
            size_t idx = (size_t)(crow + r) * DIM + ccol;
            OUT[idx] = acc[r] + bi + res[idx];
        }
    }
}

// ------------------------------ edge attention -----------------------------

// One block (128 threads, 4 waves) per edge.
// score = K[src]*Q[dst]/sqrt(16) * pe  (written back over pe, feeds O_e GEMM)
// sexp  = exp(clip(sum_dk score, -5, 5)) per head (16-lane shuffle reduction)
// wV[dst] += V[src]*sexp ; z[dst,h] += sexp   (f32 global atomics)
__global__ void edge_attn(const float* __restrict__ Q, const float* __restrict__ K,
                          const float* __restrict__ V, float* __restrict__ P,
                          const int* __restrict__ src, const int* __restrict__ dst,
                          float* __restrict__ wV, float* __restrict__ z) {
    const int f = threadIdx.x;                 // feature 0..127; head = f>>4
    const size_t eid = blockIdx.x;
    const int s = src[eid], d = dst[eid];

    float sc = K[(size_t)s * DIM + f] * Q[(size_t)d * DIM + f] * 0.25f
             * P[eid * DIM + f];
    P[eid * DIM + f] = sc;                     // out_edge_features

    float sum = sc;                            // reduce within each head (16 lanes)
    #pragma unroll
    for (int o = 8; o >= 1; o >>= 1) sum += __shfl_xor(sum, o, 16);
    float sx = expf(fminf(fmaxf(sum, -5.0f), 5.0f));

    atomicAdd(&wV[(size_t)d * DIM + f], V[(size_t)s * DIM + f] * sx);
    if ((f & 15) == 0) atomicAdd(&z[(size_t)d * NHEAD + (f >> 4)], sx);
}

// h_attn = wV / (z + 1e-6), in place on wV.
__global__ void attn_div(float* __restrict__ wV, const float* __restrict__ z, long n) {
    long i = (long)blockIdx.x * blockDim.x + threadIdx.x;
    long s = (long)gridDim.x * blockDim.x;
    for (; i < n; i += s) {
        long row  = i >> 7;
        int  head = (int)((i >> 4) & (NHEAD - 1));
        wV[i] = wV[i] / (z[row * NHEAD + head] + 1e-6f);
    }
}

// ------------------------------ driver -------------------------------------

extern "C" void kernel_launch(void* const* d_in, const int* in_sizes, int n_in,
                              void* d_out, int out_size, void* d_ws, size_t ws_size,
                              hipStream_t stream) {
    (void)in_sizes; (void)n_in; (void)out_size;
    // setup_inputs() flattening order: h, e, params{...in insertion order}, edge_src, edge_dst
    const float* h      = (const float*)d_in[0];
    const float* e      = (const float*)d_in[1];
    const float* W_Q    = (const float*)d_in[2];
    const float* W_K    = (const float*)d_in[3];
    const float* W_V    = (const float*)d_in[4];
    const float* W_pe   = (const float*)d_in[5];
    const float* O_h_w  = (const float*)d_in[6];
    const float* O_h_b  = (const float*)d_in[7];
    const float* O_e_w  = (const float*)d_in[8];
    const float* O_e_b  = (const float*)d_in[9];
    const float* bn1_hg = (const float*)d_in[10];
    const float* bn1_hb = (const float*)d_in[11];
    const float* bn1_eg = (const float*)d_in[12];
    const float* bn1_eb = (const float*)d_in[13];
    const float* bn2_hg = (const float*)d_in[14];
    const float* bn2_hb = (const float*)d_in[15];
    const float* bn2_eg = (const float*)d_in[16];
    const float* bn2_eb = (const float*)d_in[17];
    const float* F1h_w  = (const float*)d_in[18];
    const float* F1h_b  = (const float*)d_in[19];
    const float* F2h_w  = (const float*)d_in[20];
    const float* F2h_b  = (const float*)d_in[21];
    const float* F1e_w  = (const float*)d_in[22];
    const float* F1e_b  = (const float*)d_in[23];
    const float* F2e_w  = (const float*)d_in[24];
    const float* F2e_b  = (const float*)d_in[25];
    const int*   esrc   = (const int*)d_in[26];
    const int*   edst   = (const int*)d_in[27];

    float* out_h = (float*)d_out;                            // [N,128]
    float* out_e = out_h + (size_t)N_NODES * DIM;            // [E,128]

    // ---- workspace layout (floats) ----
    float* ws = (float*)d_ws;
    size_t o = 0;
    float* HN = ws + o; o += (size_t)N_NODES * DIM;          // h_in1 / h2 / h_in2
    float* EN = ws + o; o += (size_t)N_EDGES * DIM;          // e_in1 / e2 / e_in2
    float* Qb = ws + o; o += (size_t)N_NODES * DIM;
    float* Kb = ws + o; o += (size_t)N_NODES * DIM;
    float* Vb = ws + o; o += (size_t)N_NODES * DIM;
    float* Pb = ws + o; o += (size_t)N_EDGES * DIM;          // pe -> score
    float* WV = ws + o; o += (size_t)N_NODES * DIM;          // segment sums -> h_attn
    float* Zb = ws + o; o += (size_t)N_NODES * NHEAD;
    float* ST = ws + o; o += 1024;                           // 4x {sum[128],sumsq[128]}
    if (ws_size < o * sizeof(float)) return;                 // scratch too small

    float* st_h1 = ST;       float* st_e1 = ST + 256;
    float* st_h2 = ST + 512; float* st_e2 = ST + 768;

    const dim3 B256(256);
    const dim3 GS(4096);                                     // grid-stride kernels

    // ---- zero accumulators (WV, Zb, ST are contiguous) ----
    zero_f32<<<GS, B256, 0, stream>>>(WV, (long)N_NODES * DIM + (long)N_NODES * NHEAD + 1024);

    // ---- BN1 ----
    bn_stats<<<2048, DIM, 0, stream>>>(h, N_NODES, st_h1);
    bn_apply<<<GS, B256, 0, stream>>>(h, st_h1, bn1_hg, bn1_hb, HN, N_NODES);
    bn_stats<<<4096, DIM, 0, stream>>>(e, N_EDGES, st_e1);
    bn_apply<<<GS, B256, 0, stream>>>(e, st_e1, bn1_eg, bn1_eb, EN, N_EDGES);

    // ---- projections (WMMA) ----
    gemm128_wmma<<<N_NODES / 16, B256, 0, stream>>>(HN, W_Q,  nullptr, nullptr, Qb, N_NODES);
    gemm128_wmma<<<N_NODES / 16, B256, 0, stream>>>(HN, W_K,  nullptr, nullptr, Kb, N_NODES);
    gemm128_wmma<<<N_NODES / 16, B256, 0, stream>>>(HN, W_V,  nullptr, nullptr, Vb, N_NODES);
    gemm128_wmma<<<N_EDGES / 16, B256, 0, stream>>>(EN, W_pe, nullptr, nullptr, Pb, N_EDGES);

    // ---- edge attention + segment softmax-sum ----
    edge_attn<<<N_EDGES, DIM, 0, stream>>>(Qb, Kb, Vb, Pb, esrc, edst, WV, Zb);
    attn_div<<<GS, B256, 0, stream>>>(WV, Zb, (long)N_NODES * DIM);

    // ---- output projections + residual (in-place residual is thread-local safe)
    gemm128_wmma<<<N_NODES / 16, B256, 0, stream>>>(WV, O_h_w, O_h_b, HN, HN, N_NODES);
    gemm128_wmma<<<N_EDGES / 16, B256, 0, stream>>>(Pb, O_e_w, O_e_b, EN, EN, N_EDGES);

    // ---- BN2 (in place) ----
    bn_stats<<<2048, DIM, 0, stream>>>(HN, N_NODES, st_h2);
    bn_apply<<<GS, B256, 0, stream>>>(HN, st_h2, bn2_hg, bn2_hb, HN, N_NODES);
    bn_stats<<<4096, DIM, 0, stream>>>(EN, N_EDGES, st_e2);
    bn_apply<<<GS, B256, 0, stream>>>(EN, st_e2, bn2_eg, bn2_eb, EN, N_EDGES);

    // ---- fused FFN + residual -> outputs ----
    ffn_wmma<<<N_NODES / 16, B256, 0, stream>>>(HN, F1h_w, F1h_b, F2h_w, F2h_b, HN, out_h, N_NODES);
    ffn_wmma<<<N_EDGES / 16, B256, 0, stream>>>(EN, F1e_w, F1e_b, F2e_w, F2e_b, EN, out_e, N_EDGES);
}